// Het_Agg_49323404427465
// MI455X (gfx1250) — compile-verified
//
#include <hip/hip_runtime.h>
#include <math.h>

typedef __attribute__((ext_vector_type(2))) float v2f;
typedef __attribute__((ext_vector_type(8))) float v8f;

#define D 128
#define NEG_SLOPE 0.01f
#define EPS_NORM 1e-12f

// ---------------------------------------------------------------------------
// Wave reductions (wave32)
// ---------------------------------------------------------------------------
__device__ __forceinline__ float wave_sum32(float v) {
    v += __shfl_xor(v, 16, 32);
    v += __shfl_xor(v, 8, 32);
    v += __shfl_xor(v, 4, 32);
    v += __shfl_xor(v, 2, 32);
    v += __shfl_xor(v, 1, 32);
    return v;
}
// sum over the 16 lanes sharing the same half of the wave (masks < 16)
__device__ __forceinline__ float half_sum16(float v) {
    v += __shfl_xor(v, 8, 32);
    v += __shfl_xor(v, 4, 32);
    v += __shfl_xor(v, 2, 32);
    v += __shfl_xor(v, 1, 32);
    return v;
}

__device__ __forceinline__ void atom_add_f32(float* p, float v) {
    __hip_atomic_fetch_add(p, v, __ATOMIC_RELAXED, __HIP_MEMORY_SCOPE_AGENT);
}

// ---------------------------------------------------------------------------
// GEMM: H = relu(X[N,128] @ W[128,128] + bias), fp32 WMMA 16x16x4.
// W is staged in LDS once per block, K-pair packed: Wp[k/2][col] = {W[k][col],
// W[k+1][col]} so a B fragment is a single ds_load_b64.
// One wave computes a 16-row x 128-col strip: 8 v8f accumulators.
// No early returns: tail waves compute on clamped rows, skip stores
// (barrier-safe, EXEC all-1s for WMMA).
// ---------------------------------------------------------------------------
__global__ __launch_bounds__(256) void gemm_relu_f32(
    const float* __restrict__ X, const float* __restrict__ W,
    const float* __restrict__ bias, float* __restrict__ H, int nrows)
{
    __shared__ float2 Wp[(D / 2) * D];          // 64 KB

    const int lane = threadIdx.x & 31;
    const int wave = threadIdx.x >> 5;
    const int tile = blockIdx.x * 8 + wave;
    const int half = lane >> 4;                 // 0/1
    const int idx  = lane & 15;                 // 0..15
    const int m0   = tile * 16;
    const int rowA = min(m0 + idx, nrows - 1);  // clamp (no divergence)

    // Warm the A strip while we stage weights.
    __builtin_prefetch(X + (size_t)rowA * D, 0, 3);

    // Stage W into LDS, K-pair packed (coalesced along columns).
    for (int i = threadIdx.x; i < (D / 2) * D; i += 256) {
        const int k2  = i / D;
        const int col = i - k2 * D;
        float2 p;
        p.x = W[(size_t)(2 * k2    ) * D + col];
        p.y = W[(size_t)(2 * k2 + 1) * D + col];
        Wp[i] = p;
    }
    __syncthreads();

    v8f acc[8];
    #pragma unroll
    for (int nt = 0; nt < 8; ++nt) acc[nt] = (v8f){};

    for (int k0 = 0; k0 < D; k0 += 4) {
        const int kk = k0 + 2 * half;           // this lane's K pair
        const int p  = (k0 >> 1) + half;        // packed-pair row
        v2f a = *(const v2f*)(X + (size_t)rowA * D + kk);
        #pragma unroll
        for (int nt = 0; nt < 8; ++nt) {
            const int col = nt * 16 + idx;
            float2 bp = Wp[p * D + col];        // single ds_load_b64
            v2f b; b.x = bp.x; b.y = bp.y;
            acc[nt] = __builtin_amdgcn_wmma_f32_16x16x4_f32(
                false, a, false, b, (short)0, acc[nt], false, false);
        }
    }

    #pragma unroll
    for (int nt = 0; nt < 8; ++nt) {
        const int col = nt * 16 + idx;
        const float bv = bias[col];
        #pragma unroll
        for (int i = 0; i < 8; ++i) {
            const int row = m0 + i + 8 * half;  // C layout: VGPR i -> M = i + 8*half
            if (row < nrows) {
                float v = acc[nt][i] + bv;
                H[(size_t)row * D + col] = v > 0.f ? v : 0.f;
            }
        }
    }
}

// ---------------------------------------------------------------------------
// Edge aggregation: aggr[src] += h[tgt] (float4 per lane), deg[src] += 1.
// One wave (32 lanes x 4 floats = 128 cols) per edge. h/aggr are L2-resident
// (25.6 MB each << 192 MB L2), so this is L2 gather + L2 atomic bound.
// ---------------------------------------------------------------------------
__global__ __launch_bounds__(256) void edge_agg(
    const float* __restrict__ h, const int* __restrict__ edges,
    float* __restrict__ aggr, float* __restrict__ deg, int E)
{
    const long long t = (long long)blockIdx.x * blockDim.x + threadIdx.x;
    const long long e = t >> 5;
    if (e >= E) return;
    const int lane = (int)(t & 31);
    const int src = edges[e];
    const int tgt = edges[(long long)E + e];
    const int c = lane * 4;
    float4 v = *(const float4*)(h + (size_t)tgt * D + c);
    float* dst = aggr + (size_t)src * D + c;
    atom_add_f32(dst + 0, v.x);
    atom_add_f32(dst + 1, v.y);
    atom_add_f32(dst + 2, v.z);
    atom_add_f32(dst + 3, v.w);
    if (lane == 0) atom_add_f32(deg + src, 1.0f);
}

// ---------------------------------------------------------------------------
// Combine: per node, mean-normalize aggr_{r,u,b}, attention scores via u,
// softmax over 3 relations, weighted sum -> combined. One wave per node.
// ---------------------------------------------------------------------------
__global__ __launch_bounds__(256) void combine_kernel(
    const float* __restrict__ ar, const float* __restrict__ au, const float* __restrict__ ab,
    const float* __restrict__ dr, const float* __restrict__ du, const float* __restrict__ db,
    const float* __restrict__ xn, const float* __restrict__ uvec,
    float* __restrict__ cmb, int N)
{
    const long long t = (long long)blockIdx.x * blockDim.x + threadIdx.x;
    const int node = (int)(t >> 5);
    if (node >= N) return;
    const int lane = (int)(t & 31);
    const int c = lane * 4;

    float4 xv = *(const float4*)(xn + (size_t)node * D + c);
    float4 ul = *(const float4*)(uvec + c);        // u[0:128]
    float4 uh = *(const float4*)(uvec + D + c);    // u[128:256]

    float dotx = wave_sum32(xv.x * uh.x + xv.y * uh.y + xv.z * uh.z + xv.w * uh.w);

    const float ir = 1.f / fmaxf(dr[node], 1.f);
    const float iu = 1.f / fmaxf(du[node], 1.f);
    const float ib = 1.f / fmaxf(db[node], 1.f);

    float4 vr = *(const float4*)(ar + (size_t)node * D + c);
    float4 vu = *(const float4*)(au + (size_t)node * D + c);
    float4 vb = *(const float4*)(ab + (size_t)node * D + c);
    vr.x *= ir; vr.y *= ir; vr.z *= ir; vr.w *= ir;
    vu.x *= iu; vu.y *= iu; vu.z *= iu; vu.w *= iu;
    vb.x *= ib; vb.y *= ib; vb.z *= ib; vb.w *= ib;

    float tr = wave_sum32(vr.x * ul.x + vr.y * ul.y + vr.z * ul.z + vr.w * ul.w) + dotx;
    float tu = wave_sum32(vu.x * ul.x + vu.y * ul.y + vu.z * ul.z + vu.w * ul.w) + dotx;
    float tb = wave_sum32(vb.x * ul.x + vb.y * ul.y + vb.z * ul.z + vb.w * ul.w) + dotx;
    float sr = __expf(tr > 0.f ? tr : NEG_SLOPE * tr);
    float su = __expf(tu > 0.f ? tu : NEG_SLOPE * tu);
    float sb = __expf(tb > 0.f ? tb : NEG_SLOPE * tb);
    const float inv = 1.f / (sr + su + sb);
    sr *= inv; su *= inv; sb *= inv;

    float4 o;
    o.x = sr * vr.x + su * vu.x + sb * vb.x;
    o.y = sr * vr.y + su * vu.y + sb * vb.y;
    o.z = sr * vr.z + su * vu.z + sb * vb.z;
    o.w = sr * vr.w + su * vu.w + sb * vb.w;
    *(float4*)(cmb + (size_t)node * D + c) = o;
}

// ---------------------------------------------------------------------------
// Final: out = l2norm(relu(cat([x_node, combined]) @ W_lin + b_lin)).
// K = 256, staged into LDS (pair-packed) in two 64 KB chunks. All waves stay
// alive across barriers; only stores are guarded.
// ---------------------------------------------------------------------------
__global__ __launch_bounds__(256) void final_gemm_norm(
    const float* __restrict__ Xn, const float* __restrict__ Cmb,
    const float* __restrict__ Wl, const float* __restrict__ bl,
    float* __restrict__ out, int nrows)
{
    __shared__ float2 Wp[(D / 2) * D];          // 64 KB, reused for both K halves

    const int lane = threadIdx.x & 31;
    const int wave = threadIdx.x >> 5;
    const int tile = blockIdx.x * 8 + wave;
    const int half = lane >> 4;
    const int idx  = lane & 15;
    const int m0   = tile * 16;
    const int rowA = min(m0 + idx, nrows - 1);

    __builtin_prefetch(Xn  + (size_t)rowA * D, 0, 3);
    __builtin_prefetch(Cmb + (size_t)rowA * D, 0, 3);

    v8f acc[8];
    #pragma unroll
    for (int nt = 0; nt < 8; ++nt) acc[nt] = (v8f){};

    for (int kb = 0; kb < 2; ++kb) {
        __syncthreads();                        // protect LDS reuse
        // Stage this K-half of W_lin, pair-packed.
        for (int i = threadIdx.x; i < (D / 2) * D; i += 256) {
            const int k2  = i / D;
            const int col = i - k2 * D;
            const int k   = kb * D + 2 * k2;
            float2 p;
            p.x = Wl[(size_t)(k    ) * D + col];
            p.y = Wl[(size_t)(k + 1) * D + col];
            Wp[i] = p;
        }
        __syncthreads();

        const float* Asrc = (kb == 0) ? Xn : Cmb;
        for (int k0 = 0; k0 < D; k0 += 4) {
            const int kk = k0 + 2 * half;
            const int p  = (k0 >> 1) + half;
            v2f a = *(const v2f*)(Asrc + (size_t)rowA * D + kk);
            #pragma unroll
            for (int nt = 0; nt < 8; ++nt) {
                const int col = nt * 16 + idx;
                float2 bp = Wp[p * D + col];
                v2f b; b.x = bp.x; b.y = bp.y;
                acc[nt] = __builtin_amdgcn_wmma_f32_16x16x4_f32(
                    false, a, false, b, (short)0, acc[nt], false, false);
            }
        }
    }

    // bias + relu, then per-row L2 norm. Row (i, half) lives on 16 lanes (idx)
    // across the 8 column tiles.
    #pragma unroll
    for (int i = 0; i < 8; ++i) {
        const int row = m0 + i + 8 * half;
        float ss = 0.f;
        #pragma unroll
        for (int nt = 0; nt < 8; ++nt) {
            const int col = nt * 16 + idx;
            float v = acc[nt][i] + bl[col];
            v = v > 0.f ? v : 0.f;
            acc[nt][i] = v;
            ss += v * v;
        }
        ss = half_sum16(ss);
        const float scale = 1.f / fmaxf(sqrtf(ss), EPS_NORM);
        if (row < nrows) {
            #pragma unroll
            for (int nt = 0; nt < 8; ++nt)
                out[(size_t)row * D + nt * 16 + idx] = acc[nt][i] * scale;
        }
    }
}

// ---------------------------------------------------------------------------
extern "C" void kernel_launch(void* const* d_in, const int* in_sizes, int n_in,
                              void* d_out, int out_size, void* d_ws, size_t ws_size,
                              hipStream_t stream)
{
    const float* x_r    = (const float*)d_in[0];
    const float* x_u    = (const float*)d_in[1];
    const float* x_b    = (const float*)d_in[2];
    const float* x_node = (const float*)d_in[3];
    const int*   e_r    = (const int*)d_in[4];
    const int*   e_u    = (const int*)d_in[5];
    const int*   e_b    = (const int*)d_in[6];
    const float* W_r    = (const float*)d_in[8];
    const float* b_r    = (const float*)d_in[9];
    const float* W_u    = (const float*)d_in[10];
    const float* b_u    = (const float*)d_in[11];
    const float* W_b    = (const float*)d_in[12];
    const float* b_b    = (const float*)d_in[13];
    const float* uvec   = (const float*)d_in[14];
    const float* W_lin  = (const float*)d_in[15];
    const float* b_lin  = (const float*)d_in[16];

    const int N = in_sizes[0] / D;
    const int E = in_sizes[4] / 2;

    // Workspace layout (floats): h_tmp/combined | aggr_r | aggr_u | aggr_b | deg[3N]
    float* ws = (float*)d_ws;
    const size_t nd = (size_t)N * D;
    float* h_tmp = ws;
    float* aggr0 = ws + nd;            // 3 contiguous aggr buffers
    float* deg   = ws + 4 * nd;        // 3 contiguous degree buffers

    hipMemsetAsync(aggr0, 0, 3 * nd * sizeof(float), stream);
    hipMemsetAsync(deg,   0, 3 * (size_t)N * sizeof(float), stream);

    const int ntiles = (N + 15) / 16;
    const int gemmBlocks = (ntiles + 7) / 8;
    const long long ethreads = (long long)E * 32;
    const int eblocks = (int)((ethreads + 255) / 256);
    const long long cthreads = (long long)N * 32;
    const int cblocks = (int)((cthreads + 255) / 256);

    const float* xs[3] = {x_r, x_u, x_b};
    const float* Wsm[3] = {W_r, W_u, W_b};
    const float* bs[3] = {b_r, b_u, b_b};
    const int* es[3] = {e_r, e_u, e_b};

    for (int r = 0; r < 3; ++r) {
        gemm_relu_f32<<<gemmBlocks, 256, 0, stream>>>(xs[r], Wsm[r], bs[r], h_tmp, N);
        edge_agg<<<eblocks, 256, 0, stream>>>(h_tmp, es[r], aggr0 + (size_t)r * nd,
                                              deg + (size_t)r * N, E);
    }

    combine_kernel<<<cblocks, 256, 0, stream>>>(
        aggr0, aggr0 + nd, aggr0 + 2 * nd,
        deg, deg + N, deg + 2 * N,
        x_node, uvec, h_tmp, N);

    final_gemm_norm<<<gemmBlocks, 256, 0, stream>>>(
        x_node, h_tmp, W_lin, b_lin, (float*)d_out, N);
}